// Unet3DCrossLocalGeneric_57346403336679
// MI455X (gfx1250) — compile-verified
//
#include <hip/hip_runtime.h>
#include <math.h>

// Problem constants (match reference)
#define BB 1
#define CC 64
#define DD 24
#define HH 24
#define WW 24
#define NN (DD*HH*WW)      // 13824
#define KNBR (DD+HH+WW-2)  // 70
#define DS 64
#define GD 64
#define NGROUPS 16         // C/4
#define CPG 4              // channels per group
#define EPSV 1e-5f

typedef float v2f __attribute__((ext_vector_type(2)));
typedef float v8f __attribute__((ext_vector_type(8)));

// ---------------------------------------------------------------------------
// GEMM: out[n, j] = sum_c in[n, c] * w[j*64 + c] + bias[j]
//   in element [n, c] is at in[n*RS + c*CS]  (compile-time strides so the
//   compiler folds all offsets into global_load immediate fields).
//   out is (N, 64) row-major.
// One wave computes one 16-node tile across all 64 output columns with
// v_wmma_f32_16x16x4_f32 (fp32 matrix pipe). Block = 256 threads = 8 waves.
// Weights (16 KB) are staged once in LDS and shared by all 8 waves.
// ---------------------------------------------------------------------------
template<int RS, int CS>
__global__ __launch_bounds__(256) void gemm64_wmma(
    const float* __restrict__ in,
    const float* __restrict__ w,     // (64, 64) row-major: w[j*64 + c]
    const float* __restrict__ bias,  // (64)
    float* __restrict__ out)         // (N, 64) row-major
{
  __shared__ __align__(16) float s_w[64 * 64];   // 16 KB staged weights
  {
    const float4* __restrict__ src = (const float4*)w;
    float4* dst = (float4*)s_w;
    #pragma unroll
    for (int i = 0; i < 4; ++i)
      dst[threadIdx.x + i * 256] = src[threadIdx.x + i * 256];
  }
  __syncthreads();

  const int lane = threadIdx.x & 31;
  const int wave = threadIdx.x >> 5;
  const int tile = blockIdx.x * 8 + wave;     // 864 tiles total
  const int n0   = tile * 16;
  const int m    = lane & 15;                 // A row / B,D column within tile
  const int kb   = (lane >> 4) << 1;          // 0 or 2 (K sub-pair per half-wave)
  const int rb   = (lane >> 4) << 3;          // 0 or 8 (D-matrix row base)

  v8f acc0 = {0.f,0.f,0.f,0.f,0.f,0.f,0.f,0.f};
  v8f acc1 = acc0, acc2 = acc0, acc3 = acc0;

  // Per-lane bases; all per-kk offsets below are compile-time immediates.
  const float* __restrict__ inrow = in + (long)(n0 + m) * RS + (long)kb * CS;
  const float* __restrict__ w0 = &s_w[(0 * 16 + m) * 64 + kb];
  const float* __restrict__ w1 = &s_w[(1 * 16 + m) * 64 + kb];
  const float* __restrict__ w2 = &s_w[(2 * 16 + m) * 64 + kb];
  const float* __restrict__ w3 = &s_w[(3 * 16 + m) * 64 + kb];

  #pragma unroll
  for (int kk = 0; kk < 16; ++kk) {
    // A fragment: 16x4 fp32, lane holds (m, kb+kk*4) and (m, kb+kk*4+1)
    v2f a;
    a.x = inrow[(kk * 4) * CS];
    a.y = inrow[(kk * 4 + 1) * CS];
    // B fragments from LDS: B[k][j] = w[j*64 + k]; k pair is 8B contiguous
    v2f b0 = *(const v2f*)&w0[kk * 4];
    v2f b1 = *(const v2f*)&w1[kk * 4];
    v2f b2 = *(const v2f*)&w2[kk * 4];
    v2f b3 = *(const v2f*)&w3[kk * 4];
    acc0 = __builtin_amdgcn_wmma_f32_16x16x4_f32(false, a, false, b0, (short)0, acc0, false, false);
    acc1 = __builtin_amdgcn_wmma_f32_16x16x4_f32(false, a, false, b1, (short)0, acc1, false, false);
    acc2 = __builtin_amdgcn_wmma_f32_16x16x4_f32(false, a, false, b2, (short)0, acc2, false, false);
    acc3 = __builtin_amdgcn_wmma_f32_16x16x4_f32(false, a, false, b3, (short)0, acc3, false, false);
  }

  // D layout: VGPR v -> row (v + rb), column (m + t*16). Add bias, store.
  #pragma unroll
  for (int t = 0; t < 4; ++t) {
    const v8f acc = (t == 0) ? acc0 : (t == 1) ? acc1 : (t == 2) ? acc2 : acc3;
    const int j = t * 16 + m;
    const float bj = bias[j];
    #pragma unroll
    for (int v = 0; v < 8; ++v) {
      const int mm = v + rb;
      out[(long)(n0 + mm) * 64 + j] = acc[v] + bj;
    }
  }
}

// ---------------------------------------------------------------------------
// Attention: per node n:
//   f[k] = dot(theta[n], phi[nbr[n][k]]);  a = softmax(f);  y[n] = sum a[k]*g[nbr]
// Block = 256 threads = 8 waves = 8 nodes (wave32 shuffles for softmax).
// ---------------------------------------------------------------------------
__global__ __launch_bounds__(256) void attn_kernel(
    const float* __restrict__ theta,  // (N,64)
    const float* __restrict__ phi,    // (N,64)
    const float* __restrict__ gval,   // (N,64)
    const int*   __restrict__ nbr,    // (N,70)
    float* __restrict__ y)            // (N,64)
{
  __shared__ __align__(16) float s_theta[8][64];
  __shared__ float s_a[8][KNBR + 2];
  __shared__ int   s_idx[8][KNBR + 2];

  const int lane = threadIdx.x & 31;
  const int wv   = threadIdx.x >> 5;
  const int n    = blockIdx.x * 8 + wv;

  // stage theta row + neighbor indices
  s_theta[wv][lane]      = theta[(long)n * 64 + lane];
  s_theta[wv][lane + 32] = theta[(long)n * 64 + lane + 32];
  for (int k = lane; k < KNBR; k += 32) s_idx[wv][k] = nbr[(long)n * KNBR + k];
  __syncthreads();

  // logits: lane covers neighbors lane, lane+32, lane+64
  const float4* tr = (const float4*)(&s_theta[wv][0]);
  float f0 = -INFINITY, f1 = -INFINITY, f2 = -INFINITY;
  #pragma unroll
  for (int r = 0; r < 3; ++r) {
    const int k = lane + 32 * r;
    if (k < KNBR) {
      const int nb = s_idx[wv][k];
      const float4* pr = (const float4*)(phi + (long)nb * 64);
      float d = 0.f;
      #pragma unroll
      for (int q = 0; q < 16; ++q) {
        float4 t4 = tr[q];
        float4 p4 = pr[q];
        d = fmaf(t4.x, p4.x, d); d = fmaf(t4.y, p4.y, d);
        d = fmaf(t4.z, p4.z, d); d = fmaf(t4.w, p4.w, d);
      }
      if (r == 0) f0 = d; else if (r == 1) f1 = d; else f2 = d;
    }
  }

  // softmax over the 70 logits spread across the wave
  float mx = fmaxf(f0, fmaxf(f1, f2));
  #pragma unroll
  for (int off = 16; off >= 1; off >>= 1) mx = fmaxf(mx, __shfl_xor(mx, off, 32));
  const float e0 = __expf(f0 - mx);
  const float e1 = __expf(f1 - mx);
  const float e2 = (lane + 64 < KNBR) ? __expf(f2 - mx) : 0.f;
  float s = e0 + e1 + e2;
  #pragma unroll
  for (int off = 16; off >= 1; off >>= 1) s += __shfl_xor(s, off, 32);
  const float rs = 1.f / s;

  s_a[wv][lane]      = e0 * rs;
  s_a[wv][lane + 32] = e1 * rs;
  if (lane + 64 < KNBR) s_a[wv][lane + 64] = e2 * rs;
  __syncthreads();

  // weighted value sum: lane covers feature dims lane and lane+32
  float acc0 = 0.f, acc1 = 0.f;
  for (int k = 0; k < KNBR; ++k) {
    const float a  = s_a[wv][k];
    const long  nb = s_idx[wv][k];
    acc0 = fmaf(a, gval[nb * 64 + lane],      acc0);
    acc1 = fmaf(a, gval[nb * 64 + lane + 32], acc1);
  }
  y[(long)n * 64 + lane]      = acc0;
  y[(long)n * 64 + lane + 32] = acc1;
}

// ---------------------------------------------------------------------------
// GroupNorm(cross) + residual: h_out[c,n] = h_in[c,n] + GN(cross)[c,n]
// cross is (N,C); h buffers are (C,N). One block per group (16 blocks).
// ---------------------------------------------------------------------------
__global__ __launch_bounds__(256) void gn_residual(
    const float* __restrict__ cross,  // (N, C)
    const float* __restrict__ hin,    // (C, N)
    float* __restrict__ hout,         // (C, N)
    const float* __restrict__ gamma,  // (C) for this iteration
    const float* __restrict__ beta)   // (C)
{
  __shared__ float s_sum[256];
  __shared__ float s_sq[256];
  __shared__ float s_stats[2];

  const int grp   = blockIdx.x;
  const int cbase = grp * CPG;
  const int tid   = threadIdx.x;

  float sum = 0.f, sq = 0.f;
  #pragma unroll
  for (int co = 0; co < CPG; ++co) {
    const int c = cbase + co;
    for (int nn = tid; nn < NN; nn += 256) {
      const float v = cross[(long)nn * CC + c];
      sum += v; sq = fmaf(v, v, sq);
    }
  }
  s_sum[tid] = sum; s_sq[tid] = sq;
  __syncthreads();
  for (int off = 128; off >= 1; off >>= 1) {
    if (tid < off) { s_sum[tid] += s_sum[tid + off]; s_sq[tid] += s_sq[tid + off]; }
    __syncthreads();
  }
  if (tid == 0) {
    const float inv_cnt = 1.f / (float)(CPG * NN);
    const float mean = s_sum[0] * inv_cnt;
    const float var  = s_sq[0] * inv_cnt - mean * mean;
    s_stats[0] = mean;
    s_stats[1] = rsqrtf(var + EPSV);
  }
  __syncthreads();
  const float mean = s_stats[0];
  const float rinv = s_stats[1];

  #pragma unroll
  for (int co = 0; co < CPG; ++co) {
    const int c = cbase + co;
    const float ga = gamma[c], be = beta[c];
    for (int nn = tid; nn < NN; nn += 256) {
      const float v = cross[(long)nn * CC + c];
      hout[(long)c * NN + nn] = hin[(long)c * NN + nn] + (v - mean) * rinv * ga + be;
    }
  }
}

// ---------------------------------------------------------------------------
// Eval BatchNorm3d + ReLU epilogue. h is (C, N); output matches (B,C,D,H,W).
// ---------------------------------------------------------------------------
__global__ __launch_bounds__(256) void bn_relu(
    const float* __restrict__ h,
    const float* __restrict__ bn_gamma, const float* __restrict__ bn_beta,
    const float* __restrict__ bn_mean,  const float* __restrict__ bn_var,
    float* __restrict__ out)
{
  const long i = (long)blockIdx.x * blockDim.x + threadIdx.x;
  if (i >= (long)CC * NN) return;
  const int c = (int)(i / NN);
  const float v = (h[i] - bn_mean[c]) * rsqrtf(bn_var[c] + EPSV) * bn_gamma[c] + bn_beta[c];
  out[i] = v > 0.f ? v : 0.f;
}

// ---------------------------------------------------------------------------
extern "C" void kernel_launch(void* const* d_in, const int* in_sizes, int n_in,
                              void* d_out, int out_size, void* d_ws, size_t ws_size,
                              hipStream_t stream) {
  const float* x        = (const float*)d_in[0];
  const int*   nbr      = (const int*)  d_in[1];
  const float* phi_w    = (const float*)d_in[2];
  const float* phi_b    = (const float*)d_in[3];
  const float* theta_w  = (const float*)d_in[4];
  const float* theta_b  = (const float*)d_in[5];
  const float* G_w      = (const float*)d_in[6];
  const float* G_b      = (const float*)d_in[7];
  const float* r_w      = (const float*)d_in[8];
  const float* r_b      = (const float*)d_in[9];
  const float* gn_gamma = (const float*)d_in[10];
  const float* gn_beta  = (const float*)d_in[11];
  const float* bn_gamma = (const float*)d_in[12];
  const float* bn_beta  = (const float*)d_in[13];
  const float* bn_mean  = (const float*)d_in[14];
  const float* bn_var   = (const float*)d_in[15];
  float* out = (float*)d_out;

  float* ws = (float*)d_ws;
  const size_t SZ = (size_t)NN * 64;   // 884736 floats per buffer
  float* theta = ws + 0 * SZ;
  float* phi   = ws + 1 * SZ;
  float* g     = ws + 2 * SZ;
  float* y     = ws + 3 * SZ;
  float* cross = ws + 4 * SZ;
  float* h1    = ws + 5 * SZ;
  float* h2    = ws + 6 * SZ;

  const int GEMM_BLOCKS = (NN / 16) / 8;   // 864 tiles / 8 waves = 108
  const int ATTN_BLOCKS = NN / 8;          // 1728

  const float* hcur = x;
  float* hbufs[2] = { h1, h2 };
  for (int it = 0; it < 2; ++it) {
    // projections from h (C,N) layout: rs=1, cs=N (compile-time strides)
    gemm64_wmma<1, NN><<<GEMM_BLOCKS, 256, 0, stream>>>(hcur, theta_w, theta_b, theta);
    gemm64_wmma<1, NN><<<GEMM_BLOCKS, 256, 0, stream>>>(hcur, phi_w,   phi_b,   phi);
    gemm64_wmma<1, NN><<<GEMM_BLOCKS, 256, 0, stream>>>(hcur, G_w,     G_b,     g);
    // attention / softmax / weighted sum
    attn_kernel<<<ATTN_BLOCKS, 256, 0, stream>>>(theta, phi, g, nbr, y);
    // output projection from y (N,64): rs=64, cs=1 -> cross (N,C)
    gemm64_wmma<64, 1><<<GEMM_BLOCKS, 256, 0, stream>>>(y, r_w, r_b, cross);
    // group norm + residual -> next h (C,N)
    gn_residual<<<NGROUPS, 256, 0, stream>>>(cross, hcur, hbufs[it],
                                             gn_gamma + it * CC, gn_beta + it * CC);
    hcur = hbufs[it];
  }

  const long total = (long)CC * NN;
  bn_relu<<<(int)((total + 255) / 256), 256, 0, stream>>>(
      hcur, bn_gamma, bn_beta, bn_mean, bn_var, out);
}